// FarthestSamplingLayer_82454782148642
// MI455X (gfx1250) — compile-verified
//
#include <hip/hip_runtime.h>
#include <float.h>

// FPS on CDNA5 (gfx1250): one workgroup per cloud; point data staged in the
// 320KB WGP LDS as two half-wave-role-interleaved arrays:
//   sxy[2i] = {x_i, y_i}   (B rows K0,K1 -> lanes 0-15)
//   szw[2i] = {z_i, |p|^2} (B rows K2,K3 -> lanes 16-31)
// so each lane's WMMA B operand is a single aligned ds_load_b64.
// Distance pn - 2 q.p comes out of V_WMMA_F32_16X16X4_F32 with
// A = [-2qx,-2qy,-2qz,1]; |q|^2 (read free from szw[2*last+1]) is added on VALU.

typedef __attribute__((ext_vector_type(2))) float v2f;
typedef __attribute__((ext_vector_type(8))) float v8f;

#define NP 16384      // points per cloud
#define NM 8192       // samples kept per cloud
#define NB 16         // clouds
#define NF 64         // feature channels
#define NTHREADS 1024 // 32 waves (wave32)
#define CHUNKS 32     // 16-point WMMA chunks per wave: 32 waves * 32 * 16 = 16384

__global__ __launch_bounds__(NTHREADS)
void fps_kernel(const float* __restrict__ pos, int* __restrict__ glob_out) {
    extern __shared__ float smem[];
    float* sxy = smem;            // {x, y} interleaved, 128KB
    float* szw = smem + 2 * NP;   // {z, |p|^2} interleaved, 128KB
    unsigned long long* slot = (unsigned long long*)(smem + 4 * NP); // 2 slots

    const int b      = blockIdx.x;
    const int tid    = threadIdx.x;
    const int lane   = tid & 31;
    const int wave   = tid >> 5;
    const bool lo    = lane < 16;        // half-wave: K=0,1 side of A/B
    const int laneLo = lane & 15;

    // ---- stage this cloud's positions + norms into LDS (256KB) ----
    const float* pb = pos + (size_t)b * NP * 3;
    for (int i = tid; i < NP; i += NTHREADS) {
        const float xx = pb[3 * i + 0];
        const float yy = pb[3 * i + 1];
        const float zz = pb[3 * i + 2];
        sxy[2 * i + 0] = xx;
        sxy[2 * i + 1] = yy;
        szw[2 * i + 0] = zz;
        szw[2 * i + 1] = xx * xx + yy * yy + zz * zz;
    }
    if (tid == 0) { slot[0] = 0ull; slot[1] = 0ull; }
    __syncthreads();

    // Hoisted per-lane B base pointer: chunk c's operand = one b64 at bp[c*32].
    const int pBase = wave * 512 + laneLo;
    const float* bp = (lo ? sxy : szw) + 2 * pBase;

    float mind[CHUNKS];
#pragma unroll
    for (int c = 0; c < CHUNKS; ++c) mind[c] = FLT_MAX;

    int last = 0;                         // random_start=False -> index 0 first
    int* outp = glob_out + b * NM;

    for (int t = 0; t < NM; ++t) {
        if (tid == 0) outp[t] = b * NP + last;   // emit current selection

        // broadcast-selected point q: two b64 LDS broadcast reads
        const v2f qxy = *(const v2f*)(sxy + 2 * last);   // {qx, qy}
        const v2f qzw = *(const v2f*)(szw + 2 * last);   // {qz, |q|^2}
        const float qn = qzw.y;

        // A (16x4, rows identical): lanes 0-15 -> K0,K1 ; lanes 16-31 -> K2,K3
        v2f a;
        a.x = lo ? (-2.0f * qxy.x) : (-2.0f * qzw.x);
        a.y = lo ? (-2.0f * qxy.y) : 1.0f;

        float bestVal = -1.0f;
        int   bestIdx = 0;
#pragma unroll
        for (int c = 0; c < CHUNKS; ++c) {
            const v2f bm = *(const v2f*)(bp + c * 32);   // one ds_load_b64
            v8f cz = {};
            // D[i][j] = -2 q.p_j + |p_j|^2 (rows replicated across lanes)
            v8f d = __builtin_amdgcn_wmma_f32_16x16x4_f32(
                false, a, false, bm, (short)0, cz, false, false);
            const float dist = d[0] + qn;        // |p_j - q|^2
            const float nm = fminf(mind[c], dist);
            mind[c] = nm;
            if (nm > bestVal) { bestVal = nm; bestIdx = pBase + c * 16; }
        }

        // ---- wave32 argmax (ties -> smallest index, matching jnp.argmax) ----
#pragma unroll
        for (int off = 16; off > 0; off >>= 1) {
            const float ov = __shfl_xor(bestVal, off, 32);
            const int   oi = __shfl_xor(bestIdx, off, 32);
            if (ov > bestVal || (ov == bestVal && oi < bestIdx)) {
                bestVal = ov; bestIdx = oi;
            }
        }

        // ---- cross-wave argmax: packed key, one ds_max_u64 per wave ----
        const int S = t & 1;
        if (lane == 0) {
            const unsigned long long pk =
                ((unsigned long long)__float_as_uint(bestVal) << 32) |
                (unsigned long long)(0xFFFFFFFFu - (unsigned)bestIdx);
            atomicMax(&slot[S], pk);
        }
        __syncthreads();
        const unsigned long long win = slot[S];
        last = (int)(0xFFFFFFFFu - (unsigned)(win & 0xFFFFFFFFull));
        if (tid == 0) slot[1 - S] = 0ull;   // prepare other buffer for t+1
        __syncthreads();
    }
}

// Gather batch/x/pos at the selected global indices. 16 threads per row copy
// the 64-float feature row as float4s; lane 0 of each row handles pos + batch.
__global__ __launch_bounds__(256)
void gather_kernel(const float* __restrict__ x,
                   const float* __restrict__ pos,
                   const int*   __restrict__ batch,
                   const int*   __restrict__ glob,
                   int*   __restrict__ batch_out,
                   float* __restrict__ x_out,
                   float* __restrict__ pos_out) {
    const int t   = blockIdx.x * 256 + threadIdx.x;
    const int row = t >> 4;
    const int j   = t & 15;
    if (row >= NB * NM) return;
    const int g = glob[row];
    const float4* src = (const float4*)(x + (size_t)g * NF);
    float4*       dst = (float4*)(x_out + (size_t)row * NF);
    dst[j] = src[j];
    if (j == 0) {
        batch_out[row]       = batch[g];
        pos_out[row * 3 + 0] = pos[(size_t)g * 3 + 0];
        pos_out[row * 3 + 1] = pos[(size_t)g * 3 + 1];
        pos_out[row * 3 + 2] = pos[(size_t)g * 3 + 2];
    }
}

extern "C" void kernel_launch(void* const* d_in, const int* in_sizes, int n_in,
                              void* d_out, int out_size, void* d_ws, size_t ws_size,
                              hipStream_t stream) {
    const float* pos   = (const float*)d_in[0];   // [B*P, 3] f32
    const float* x     = (const float*)d_in[1];   // [B*P, 64] f32
    const int*   batch = (const int*)d_in[2];     // [B*P] i32

    int* glob = (int*)d_ws;                       // [B*M] selected flat indices

    const size_t smem = (size_t)4 * NP * sizeof(float) + 2 * sizeof(unsigned long long);
    (void)hipFuncSetAttribute((const void*)fps_kernel,
                              hipFuncAttributeMaxDynamicSharedMemorySize, (int)smem);
    fps_kernel<<<NB, NTHREADS, smem, stream>>>(pos, glob);

    // d_out = concat(batch_out i32[B*M], x_out f32[B*M*64], pos_out f32[B*M*3])
    int*   batch_out = (int*)d_out;
    float* x_out     = (float*)d_out + (size_t)NB * NM;
    float* pos_out   = x_out + (size_t)NB * NM * NF;

    const int rows = NB * NM;
    gather_kernel<<<(rows * 16 + 255) / 256, 256, 0, stream>>>(
        x, pos, batch, glob, batch_out, x_out, pos_out);
}